// HSSCLoss_26147760898518
// MI455X (gfx1250) — compile-verified
//
#include <hip/hip_runtime.h>
#include <hip/hip_bf16.h>
#include <math.h>

// ---------------- problem constants ----------------
#define N_ROWS 16384
#define D_FEAT 1000
#define C_CLS  1000
#define CPAD   1024
#define DPAD   1024
#define KSPLIT 4
#define KCHUNK (N_ROWS / KSPLIT)          // 4096
#define NCHUNK 64
#define ROWS_PER_CHUNK (N_ROWS / NCHUNK)  // 256

typedef __attribute__((ext_vector_type(2))) float v2f;
typedef __attribute__((ext_vector_type(8))) float v8f;

// ---------------- workspace layout (float offsets) ----------------
#define WS_SUMSPART 0u                                        // KSPLIT * CPAD*DPAD
#define WS_SUMS     ((size_t)KSPLIT * CPAD * DPAD)            // CPAD*DPAD
#define WS_COUNTS   (WS_SUMS + (size_t)CPAD * DPAD)           // CPAD ints
#define WS_COLS     (WS_COUNTS + CPAD)                        // DPAD  (column sums S[d])
#define WS_COLN     (WS_COLS + DPAD)                          // DPAD  (column norms)
#define WS_COLPART  (WS_COLN + DPAD)                          // 2 * NCHUNK * DPAD
#define WS_LOSS     (WS_COLPART + 2 * NCHUNK * DPAD)          // N_ROWS

// ---------------- init: zero class counts ----------------
__global__ void zero_counts_kernel(float* __restrict__ ws) {
  int t = blockIdx.x * blockDim.x + threadIdx.x;
  if (t < CPAD) ((int*)(ws + WS_COUNTS))[t] = 0;
}

// ---------------- class counts (integer atomics: deterministic) -------------
__global__ void count_kernel(const int* __restrict__ labels, float* __restrict__ ws) {
  int i = blockIdx.x * blockDim.x + threadIdx.x;
  if (i < N_ROWS) atomicAdd(&((int*)(ws + WS_COUNTS))[labels[i]], 1);
}

// ---------------- column partial sums: S[d], Q[d] (coalesced over d) --------
__global__ void __launch_bounds__(256)
col_stats_kernel(const float* __restrict__ F, float* __restrict__ ws) {
  int d = blockIdx.x * blockDim.x + threadIdx.x;   // 0..1023
  int chunk = blockIdx.y;                          // 0..63
  float s = 0.f, q = 0.f;
  if (d < D_FEAT) {
    const float* p = F + (size_t)chunk * ROWS_PER_CHUNK * D_FEAT + d;
    for (int r = 0; r < ROWS_PER_CHUNK; ++r) {
      float f = p[(size_t)r * D_FEAT];
      s += f; q += f * f;
    }
  }
  ws[WS_COLPART + (size_t)chunk * DPAD + d] = s;
  ws[WS_COLPART + (size_t)NCHUNK * DPAD + (size_t)chunk * DPAD + d] = q;
}

// ---------------- deterministic column reduce ----------------
__global__ void __launch_bounds__(256)
col_reduce_kernel(float* __restrict__ ws) {
  int d = blockIdx.x * blockDim.x + threadIdx.x;   // 0..1023
  float s = 0.f, q = 0.f;
  for (int c = 0; c < NCHUNK; ++c) {
    s += ws[WS_COLPART + (size_t)c * DPAD + d];
    q += ws[WS_COLPART + (size_t)NCHUNK * DPAD + (size_t)c * DPAD + d];
  }
  ws[WS_COLS + d] = s;
  ws[WS_COLN + d] = sqrtf(q);
}

// ---------------- segment-sum as one-hot fp32 GEMM on WMMA ----------------
// sumsPart[part] = OneHot(labels[kpart])^T (1024x4096) @ F[kpart] (4096x1024)
// One wave per block; 8 class-tiles (8 accumulators) per wave for B-fragment
// reuse. V_WMMA_F32_16X16X4_F32, K stepped by 4.
__global__ void __launch_bounds__(32)
segsum_wmma_kernel(const float* __restrict__ F, const int* __restrict__ labels,
                   float* __restrict__ ws) {
  __shared__ int slab[KCHUNK];                     // 16 KB of this WGP's 320 KB LDS
  const int lane   = threadIdx.x;                  // 0..31 (wave32)
  const int dTile  = blockIdx.x;                   // 0..63  -> 16 feature cols
  const int cGroup = blockIdx.y;                   // 0..7   -> 8 class tiles (128 classes)
  const int part   = blockIdx.z;                   // 0..KSPLIT-1
  const int k0base = part * KCHUNK;

  // stage this K-partition's labels into LDS once
  for (int j = lane; j < KCHUNK; j += 32) slab[j] = labels[k0base + j];
  __syncthreads();

  const int  dcol  = dTile * 16 + (lane & 15);
  const int  colc  = (dcol < D_FEAT) ? dcol : 0;           // branchless pad guard
  const float colw = (dcol < D_FEAT) ? 1.f : 0.f;
  const int  khalf = (lane >> 4) << 1;                     // 0 (lanes 0-15) or 2
  const int  myc0  = cGroup * 128 + (lane & 15);           // class of tile t = myc0 + 16t

  v8f acc[8] = {};                                         // 8 x 16x16 f32 accumulators

  for (int kk = 0; kk < KCHUNK; kk += 4) {
    const int krow = kk + khalf;
    const int lab0 = slab[krow];                           // LDS broadcast (same addr/half)
    const int lab1 = slab[krow + 1];
    // B fragment: 4x16 f32; VGPR0 = K(khalf), VGPR1 = K(khalf+1), coalesced 64B/half
    const size_t base = (size_t)(k0base + krow) * D_FEAT + colc;
    v2f b;
    b.x = F[base] * colw;
    b.y = F[base + D_FEAT] * colw;
#pragma unroll
    for (int t = 0; t < 8; ++t) {
      const int c = myc0 + t * 16;
      v2f a;                                               // exact fp32 one-hot
      a.x = (lab0 == c) ? 1.f : 0.f;
      a.y = (lab1 == c) ? 1.f : 0.f;
      acc[t] = __builtin_amdgcn_wmma_f32_16x16x4_f32(
          /*neg_a=*/false, a, /*neg_b=*/false, b,
          /*c_mod=*/(short)0, acc[t], /*reuse_a=*/false, /*reuse_b=*/false);
    }
  }

  // store: C/D layout -> VGPR r: lanes 0-15 = row r, lanes 16-31 = row 8+r
  float* out = ws + WS_SUMSPART + (size_t)part * CPAD * DPAD;
  const int nCol = dTile * 16 + (lane & 15);
#pragma unroll
  for (int t = 0; t < 8; ++t) {
    const int cBase = cGroup * 128 + t * 16 + (lane >> 4) * 8;
#pragma unroll
    for (int r = 0; r < 8; ++r) {
      out[(size_t)(cBase + r) * DPAD + nCol] = acc[t][r];
    }
  }
}

// ---------------- deterministic reduce of K-partition partials --------------
__global__ void __launch_bounds__(256)
segsum_reduce_kernel(float* __restrict__ ws) {
  size_t idx = (size_t)blockIdx.x * blockDim.x + threadIdx.x;  // 1M elements
  if (idx < (size_t)CPAD * DPAD) {
    float s = 0.f;
    for (int p = 0; p < KSPLIT; ++p)
      s += ws[WS_SUMSPART + (size_t)p * CPAD * DPAD + idx];
    ws[WS_SUMS + idx] = s;
  }
}

// ---------------- fused per-row pass: cos_i, denom, gate, loss --------------
__global__ void __launch_bounds__(256)
row_kernel(const float* __restrict__ F, const int* __restrict__ labels,
           float* __restrict__ ws) {
  const int i   = blockIdx.x;
  const int tid = threadIdx.x;
  const int l   = labels[i];
  const float cnt = (float)((const int*)(ws + WS_COUNTS))[l];
  const float* __restrict__ sums = ws + WS_SUMS + (size_t)l * DPAD;  // gathered class sum
  const float* __restrict__ S    = ws + WS_COLS;
  const float* __restrict__ CN   = ws + WS_COLN;
  const float* __restrict__ f    = F + (size_t)i * D_FEAT;
  const float sqrtN = sqrtf((float)N_ROWS);

  float dot = 0.f, f2 = 0.f, c2 = 0.f, den = 0.f, mx = -1e30f;
  for (int d = tid; d < D_FEAT; d += 256) {
    const float fv = f[d];
    const float sv = sums[d];
    dot += fv * sv;
    f2  += fv * fv;
    c2  += sv * sv;
    const float num = fv * S[d];
    const float dd  = fmaxf(sqrtN * fabsf(fv) * CN[d], 1e-6f);
    den += expf(num / dd);                      // TAU = 1
    mx = fmaxf(mx, fv);
  }
  __shared__ float sd[256], sf[256], sc[256], se[256], sm[256];
  sd[tid] = dot; sf[tid] = f2; sc[tid] = c2; se[tid] = den; sm[tid] = mx;
  __syncthreads();
  for (int s = 128; s > 0; s >>= 1) {           // deterministic tree reduce
    if (tid < s) {
      sd[tid] += sd[tid + s]; sf[tid] += sf[tid + s]; sc[tid] += sc[tid + s];
      se[tid] += se[tid + s]; sm[tid] = fmaxf(sm[tid], sm[tid + s]);
    }
    __syncthreads();
  }
  if (tid == 0) {
    float cos_i = 0.f;
    if (cnt > 0.f) {
      const float dotc = sd[0] / cnt;           // dot(f, center)
      const float nf   = sqrtf(sf[0]);
      const float nc   = sqrtf(sc[0]) / cnt;    // |center|
      cos_i = dotc / fmaxf(nf * nc, 1e-6f);
    }                                           // empty class: center=0 -> cos=0
    const bool mask = (sm[0] > 0.9f) && (cos_i > 0.1f);
    ws[WS_LOSS + i] = mask ? -(cos_i - logf(se[0])) : 0.f;
  }
}

// ---------------- deterministic final mean ----------------
__global__ void __launch_bounds__(256)
final_reduce_kernel(const float* __restrict__ ws, float* __restrict__ out) {
  __shared__ float sh[256];
  float s = 0.f;
  for (int i = threadIdx.x; i < N_ROWS; i += 256) s += ws[WS_LOSS + i];
  sh[threadIdx.x] = s;
  __syncthreads();
  for (int t = 128; t > 0; t >>= 1) {
    if (threadIdx.x < t) sh[threadIdx.x] += sh[threadIdx.x + t];
    __syncthreads();
  }
  if (threadIdx.x == 0) out[0] = sh[0] / (float)N_ROWS;
}

extern "C" void kernel_launch(void* const* d_in, const int* in_sizes, int n_in,
                              void* d_out, int out_size, void* d_ws, size_t ws_size,
                              hipStream_t stream) {
  const float* F      = (const float*)d_in[0];
  const int*   labels = (const int*)d_in[1];
  float*       out    = (float*)d_out;
  float*       ws     = (float*)d_ws;
  (void)in_sizes; (void)n_in; (void)out_size; (void)ws_size;

  zero_counts_kernel<<<dim3((CPAD + 255) / 256), dim3(256), 0, stream>>>(ws);
  count_kernel<<<dim3((N_ROWS + 255) / 256), dim3(256), 0, stream>>>(labels, ws);

  col_stats_kernel<<<dim3(DPAD / 256, NCHUNK), dim3(256), 0, stream>>>(F, ws);
  col_reduce_kernel<<<dim3(DPAD / 256), dim3(256), 0, stream>>>(ws);

  // one-hot fp32 GEMM segment-sum: 64 dTiles x 8 cGroups x KSPLIT waves
  segsum_wmma_kernel<<<dim3(DPAD / 16, CPAD / 128, KSPLIT), dim3(32), 0, stream>>>(
      F, labels, ws);
  segsum_reduce_kernel<<<dim3((CPAD * DPAD + 255) / 256), dim3(256), 0, stream>>>(ws);

  row_kernel<<<dim3(N_ROWS), dim3(256), 0, stream>>>(F, labels, ws);
  final_reduce_kernel<<<dim3(1), dim3(256), 0, stream>>>(ws, out);
}